// MaceLayer_79508434583946
// MI455X (gfx1250) — compile-verified
//
#include <hip/hip_runtime.h>
#include <math.h>

// ---------------------------------------------------------------------------
// MACE layer for MI455X (gfx1250, wave32). fp32 WMMA (v_wmma_f32_16x16x4_f32)
// for all GEMMs; fused edge kernel (radial MLP + messages + atomic scatter).
// ---------------------------------------------------------------------------

typedef __attribute__((ext_vector_type(2))) float v2f;
typedef __attribute__((ext_vector_type(8))) float v8f;

#define C_CH   64
#define R_DIM  8
#define HR_DIM 64
#define HEADS  2
#define INV_AVG (1.0f / 16.0f)
#define SOFT_N  2.0f

// ---------------------------------------------------------------------------
// Generic fp32 WMMA GEMM: Y[M,Ncols] = X[M,K] @ W[K,Ncols]
// X/Y element (z,row,col) at  base + z*zoff + row*ld + col*stride
// One wave computes one 16x16 tile. Requires M % 16 == 0, K % 4 == 0,
// Ncols % 16 == 0. EXEC all-ones at every WMMA (uniform guard only).
// ---------------------------------------------------------------------------
#define GWAVES 4
__global__ __launch_bounds__(32 * GWAVES)
void gemm_wmma_f32(const float* __restrict__ X, int ldx, int xstride, int zoffx,
                   const float* __restrict__ W, int Ncols,
                   float* __restrict__ Y, int ldy, int ystride, int zoffy,
                   int M, int K)
{
    const int lane = threadIdx.x & 31;
    const int wv   = threadIdx.x >> 5;
    const int row0 = (blockIdx.y * GWAVES + wv) * 16;
    if (row0 >= M) return;
    const int n0   = blockIdx.x * 16;
    const int half = lane >> 4;        // 0: K pair {0,1}; 1: K pair {2,3}
    const int q    = lane & 15;

    const float* Xz = X + (size_t)blockIdx.z * zoffx;
    float*       Yz = Y + (size_t)blockIdx.z * zoffy;

    v8f acc = {0.f, 0.f, 0.f, 0.f, 0.f, 0.f, 0.f, 0.f};

    const int ksteps = K >> 2;
    for (int kk = 0; kk < ksteps; ++kk) {
        const int k0 = 4 * kk + 2 * half;
        v2f a, b;
        // A 16x4 f32 layout: lanes0-15 M=lane K={0,1}; lanes16-31 M=lane-16 K={2,3}
        a.x = Xz[(size_t)(row0 + q) * ldx + (size_t)k0 * xstride];
        a.y = Xz[(size_t)(row0 + q) * ldx + (size_t)(k0 + 1) * xstride];
        // B 4x16 f32 layout (mirror of A): lanes0-15 N=lane K={0,1}; lanes16-31 K={2,3}
        b.x = W[(size_t)k0 * Ncols + n0 + q];
        b.y = W[(size_t)(k0 + 1) * Ncols + n0 + q];
        acc = __builtin_amdgcn_wmma_f32_16x16x4_f32(false, a, false, b,
                                                    (short)0, acc, false, false);
    }
    // C/D layout: VGPR r -> row r (lanes0-15) / r+8 (lanes16-31), col = lane&15
#pragma unroll
    for (int r = 0; r < 8; ++r) {
        const int row = row0 + r + 8 * half;
        Yz[(size_t)row * ldy + (size_t)(n0 + q) * ystride] = acc[r];
    }
}

// ---------------------------------------------------------------------------
// Residual: res_s[n,d] = sum_c fs[n,c] * W_rs[sp,c,d]   (and per-component v)
// ---------------------------------------------------------------------------
__global__ __launch_bounds__(256)
void resid_kernel(const float* __restrict__ fs, const float* __restrict__ fv,
                  const int* __restrict__ species,
                  const float* __restrict__ W_rs, const float* __restrict__ W_rv,
                  float* __restrict__ res_s, float* __restrict__ res_v, int N)
{
    const int d = threadIdx.x;                       // 0..63
    const int n = blockIdx.x * blockDim.y + threadIdx.y;
    if (n >= N) return;
    const int sp = species[n];
    const float* Ws = W_rs + (size_t)sp * C_CH * C_CH;
    const float* Wv = W_rv + (size_t)sp * C_CH * C_CH;
    float as = 0.f, a0 = 0.f, a1 = 0.f, a2 = 0.f;
    for (int c = 0; c < C_CH; ++c) {
        const float ws = Ws[c * C_CH + d];
        const float wvv = Wv[c * C_CH + d];
        as += fs[(size_t)n * C_CH + c] * ws;
        const size_t vi = ((size_t)n * C_CH + c) * 3;
        a0 += fv[vi + 0] * wvv;
        a1 += fv[vi + 1] * wvv;
        a2 += fv[vi + 2] * wvv;
    }
    res_s[(size_t)n * C_CH + d] = as;
    const size_t vo = ((size_t)n * C_CH + d) * 3;
    res_v[vo + 0] = a0; res_v[vo + 1] = a1; res_v[vo + 2] = a2;
}

// ---------------------------------------------------------------------------
// Fused edge kernel: per wave, 16 edges.
//   A) u = normalize(edge_vec); h = silu(remb @ Wr1)            -> LDS
//   B) w = h @ Wr2  via 20x16 fp32 WMMAs (Wr2 cache-resident)   -> LDS
//   C) messages (dot/cross/5-path mix) + atomic scatter (scaled by 1/AVG)
// ---------------------------------------------------------------------------
#define EPW 16
#define EWAVES 2
__global__ __launch_bounds__(32 * EWAVES)
void edge_kernel(const float* __restrict__ evec, const float* __restrict__ remb,
                 const int* __restrict__ senders, const int* __restrict__ receivers,
                 const float* __restrict__ Wr1, const float* __restrict__ Wr2,
                 const float* __restrict__ s_u, const float* __restrict__ v_u,
                 float* __restrict__ a_s, float* __restrict__ a_v, int E)
{
    __shared__ float htile[EWAVES][EPW][HR_DIM + 2];   // +2 pad: avoid bank conflicts
    __shared__ float wtile[EWAVES][EPW][5 * C_CH];
    __shared__ float utile[EWAVES][EPW][4];

    const int lane = threadIdx.x & 31;
    const int wv   = threadIdx.x >> 5;
    const int e0   = (blockIdx.x * EWAVES + wv) * EPW;
    const bool act = (e0 < E);

    if (act) {
        // --- phase A: directions + hidden MLP ---
        if (lane < EPW) {
            const int e = e0 + lane;
            const float ex = evec[(size_t)e * 3 + 0];
            const float ey = evec[(size_t)e * 3 + 1];
            const float ez = evec[(size_t)e * 3 + 2];
            const float inv = 1.0f / fmaxf(sqrtf(ex * ex + ey * ey + ez * ez), 1e-9f);
            utile[wv][lane][0] = ex * inv;
            utile[wv][lane][1] = ey * inv;
            utile[wv][lane][2] = ez * inv;
        }
        for (int idx = lane; idx < EPW * HR_DIM; idx += 32) {
            const int m = idx >> 6;          // edge within tile
            const int j = idx & (HR_DIM - 1);
            float acc = 0.f;
#pragma unroll
            for (int r = 0; r < R_DIM; ++r)
                acc += remb[(size_t)(e0 + m) * R_DIM + r] * Wr1[r * HR_DIM + j];
            htile[wv][m][j] = acc / (1.0f + expf(-acc));   // silu
        }
    }
    __syncthreads();

    if (act) {
        // --- phase B: w = h @ Wr2  (16 x 64 @ 64 x 320), fp32 WMMA ---
        const int half = lane >> 4;
        const int q    = lane & 15;
        v2f afrag[16];
#pragma unroll
        for (int kk = 0; kk < 16; ++kk) {
            afrag[kk].x = htile[wv][q][4 * kk + 2 * half];
            afrag[kk].y = htile[wv][q][4 * kk + 2 * half + 1];
        }
        for (int nt = 0; nt < 20; ++nt) {
            v8f acc = {0.f, 0.f, 0.f, 0.f, 0.f, 0.f, 0.f, 0.f};
#pragma unroll
            for (int kk = 0; kk < 16; ++kk) {
                const int k0 = 4 * kk + 2 * half;
                v2f b;
                b.x = Wr2[(size_t)k0 * 320 + nt * 16 + q];
                b.y = Wr2[(size_t)(k0 + 1) * 320 + nt * 16 + q];
                acc = __builtin_amdgcn_wmma_f32_16x16x4_f32(false, afrag[kk], false, b,
                                                            (short)0, acc, false, false);
            }
#pragma unroll
            for (int r = 0; r < 8; ++r)
                wtile[wv][r + 8 * half][nt * 16 + q] = acc[r];
        }
    }
    __syncthreads();

    if (act) {
        // --- phase C: messages + scatter ---
        for (int m = 0; m < EPW; ++m) {
            const int e   = e0 + m;
            const int snd = senders[e];
            const int rcv = receivers[e];
            const float ux = utile[wv][m][0];
            const float uy = utile[wv][m][1];
            const float uz = utile[wv][m][2];
#pragma unroll
            for (int ci = 0; ci < 2; ++ci) {
                const int c = lane + 32 * ci;
                const size_t si = (size_t)snd * C_CH + c;
                const float se = s_u[si];
                const float v0 = v_u[si * 3 + 0];
                const float v1 = v_u[si * 3 + 1];
                const float v2 = v_u[si * 3 + 2];
                const float w0 = wtile[wv][m][0 * C_CH + c];
                const float w1 = wtile[wv][m][1 * C_CH + c];
                const float w2 = wtile[wv][m][2 * C_CH + c];
                const float w3 = wtile[wv][m][3 * C_CH + c];
                const float w4 = wtile[wv][m][4 * C_CH + c];
                const float dot = v0 * ux + v1 * uy + v2 * uz;
                const float cx = v1 * uz - v2 * uy;
                const float cy = v2 * ux - v0 * uz;
                const float cz = v0 * uy - v1 * ux;
                const size_t ri = (size_t)rcv * C_CH + c;
                atomicAdd(&a_s[ri], (w0 * se + w3 * dot) * INV_AVG);
                atomicAdd(&a_v[ri * 3 + 0], (w1 * v0 + w2 * ux + w4 * cx) * INV_AVG);
                atomicAdd(&a_v[ri * 3 + 1], (w1 * v1 + w2 * uy + w4 * cy) * INV_AVG);
                atomicAdd(&a_v[ri * 3 + 2], (w1 * v2 + w2 * uz + w4 * cz) * INV_AVG);
            }
        }
    }
}

// ---------------------------------------------------------------------------
// Product basis (correlation <= 2), species-indexed per-channel weights.
// ---------------------------------------------------------------------------
__global__ __launch_bounds__(256)
void prod_elem(const float* __restrict__ d_s, const float* __restrict__ d_v,
               const int* __restrict__ species,
               const float* __restrict__ w1s, const float* __restrict__ w1v,
               const float* __restrict__ w2ss, const float* __restrict__ w2vv,
               const float* __restrict__ w2sv,
               float* __restrict__ p_s, float* __restrict__ p_v, int N)
{
    const int i = blockIdx.x * blockDim.x + threadIdx.x;   // n*64 + c
    if (i >= N * C_CH) return;
    const int n = i >> 6;
    const int c = i & (C_CH - 1);
    const int sp = species[n];
    const int wi = sp * C_CH + c;
    const float as = d_s[i];
    const float v0 = d_v[(size_t)i * 3 + 0];
    const float v1 = d_v[(size_t)i * 3 + 1];
    const float v2 = d_v[(size_t)i * 3 + 2];
    const float vv = v0 * v0 + v1 * v1 + v2 * v2;
    p_s[i] = w1s[wi] * as + w2ss[wi] * as * as + w2vv[wi] * vv;
    const float k = w1v[wi] + w2sv[wi] * as;
    p_v[(size_t)i * 3 + 0] = k * v0;
    p_v[(size_t)i * 3 + 1] = k * v1;
    p_v[(size_t)i * 3 + 2] = k * v2;
}

// ---------------------------------------------------------------------------
// Soft-norm + residual + readout heads.
// ---------------------------------------------------------------------------
__device__ __forceinline__ float phi_soft(float nrm)
{
    const float x = nrm * (1.0f / SOFT_N);
    const float s = (x > 0.0f) ? expf(-1.0f / x) : 0.0f;
    return 1.0f / (1.0f + x * s);
}

__global__ __launch_bounds__(256)
void final_kernel(const float* __restrict__ q_s, const float* __restrict__ q_v,
                  const float* __restrict__ res_s, const float* __restrict__ res_v,
                  const float* __restrict__ w_read,
                  float* __restrict__ node_out, float* __restrict__ f_s,
                  float* __restrict__ f_v, int N)
{
    __shared__ float fs_sh[4][C_CH];
    const int c = threadIdx.x;
    const int n = blockIdx.x * 4 + threadIdx.y;
    if (n < N) {
        const size_t i = (size_t)n * C_CH + c;
        const float ps = q_s[i];
        const float fs = ps * phi_soft(fabsf(ps)) + res_s[i];
        const float v0 = q_v[i * 3 + 0];
        const float v1 = q_v[i * 3 + 1];
        const float v2 = q_v[i * 3 + 2];
        const float pv = phi_soft(sqrtf(v0 * v0 + v1 * v1 + v2 * v2));
        f_s[i] = fs;
        f_v[i * 3 + 0] = v0 * pv + res_v[i * 3 + 0];
        f_v[i * 3 + 1] = v1 * pv + res_v[i * 3 + 1];
        f_v[i * 3 + 2] = v2 * pv + res_v[i * 3 + 2];
        fs_sh[threadIdx.y][c] = fs;
    }
    __syncthreads();
    if (n < N && c < HEADS) {
        float acc = 0.f;
#pragma unroll
        for (int cc = 0; cc < C_CH; ++cc)
            acc += fs_sh[threadIdx.y][cc] * w_read[c * C_CH + cc];
        node_out[(size_t)n * HEADS + c] = acc;
    }
}

// ---------------------------------------------------------------------------
// Host launch
// ---------------------------------------------------------------------------
extern "C" void kernel_launch(void* const* d_in, const int* in_sizes, int n_in,
                              void* d_out, int out_size, void* d_ws, size_t ws_size,
                              hipStream_t stream)
{
    const float* edge_vectors = (const float*)d_in[0];
    const float* node_feats_s = (const float*)d_in[1];
    const float* node_feats_v = (const float*)d_in[2];
    const float* radial_emb   = (const float*)d_in[3];
    const int*   node_species = (const int*)d_in[4];
    const int*   senders      = (const int*)d_in[5];
    const int*   receivers    = (const int*)d_in[6];
    const float* W_rs    = (const float*)d_in[7];
    const float* W_rv    = (const float*)d_in[8];
    const float* W_up_s  = (const float*)d_in[9];
    const float* W_up_v  = (const float*)d_in[10];
    const float* Wr1     = (const float*)d_in[11];
    const float* Wr2     = (const float*)d_in[12];
    const float* W_down_s = (const float*)d_in[13];
    const float* W_down_v = (const float*)d_in[14];
    const float* w1s  = (const float*)d_in[15];
    const float* w1v  = (const float*)d_in[16];
    const float* w2ss = (const float*)d_in[17];
    const float* w2vv = (const float*)d_in[18];
    const float* w2sv = (const float*)d_in[19];
    const float* W_prod_s = (const float*)d_in[20];
    const float* W_prod_v = (const float*)d_in[21];
    const float* w_read   = (const float*)d_in[22];

    const int N = in_sizes[1] / C_CH;
    const int E = in_sizes[5];
    const size_t NC = (size_t)N * C_CH;

    // workspace layout (floats): res_s | res_v | s_u | v_u | a_s | a_v  (12*NC)
    float* ws    = (float*)d_ws;
    float* res_s = ws;
    float* res_v = res_s + NC;
    float* s_u   = res_v + 3 * NC;
    float* v_u   = s_u + NC;
    float* a_s   = v_u + 3 * NC;
    float* a_v   = a_s + NC;
    // lifetimes allow reuse:
    float* dn_s = s_u;  float* dn_v = v_u;   // down outputs (s_u/v_u dead)
    float* p_s  = a_s;  float* p_v  = a_v;   // product-basis (a_s/a_v dead)
    float* q_s  = s_u;  float* q_v  = v_u;   // prod-GEMM outputs (dn dead)

    float* out_node = (float*)d_out;
    float* out_fs   = out_node + (size_t)N * HEADS;
    float* out_fv   = out_fs + NC;

    const int mtiles = N / 16;                         // N % 16 == 0 (20000)
    const dim3 ggrid(C_CH / 16, (mtiles + GWAVES - 1) / GWAVES, 1);
    const dim3 ggridv(C_CH / 16, (mtiles + GWAVES - 1) / GWAVES, 3);
    const dim3 gblk(32 * GWAVES);

    // zero accumulators (a_s|a_v contiguous)
    hipMemsetAsync(a_s, 0, 4 * NC * sizeof(float), stream);

    // residual
    resid_kernel<<<N / 4, dim3(C_CH, 4), 0, stream>>>(
        node_feats_s, node_feats_v, node_species, W_rs, W_rv, res_s, res_v, N);

    // linear up
    gemm_wmma_f32<<<ggrid, gblk, 0, stream>>>(node_feats_s, C_CH, 1, 0,
                                              W_up_s, C_CH, s_u, C_CH, 1, 0, N, C_CH);
    gemm_wmma_f32<<<ggridv, gblk, 0, stream>>>(node_feats_v, 3 * C_CH, 3, 1,
                                               W_up_v, C_CH, v_u, 3 * C_CH, 3, 1, N, C_CH);

    // fused edge pipeline (radial MLP + messages + scatter)
    edge_kernel<<<(E + EPW * EWAVES - 1) / (EPW * EWAVES), 32 * EWAVES, 0, stream>>>(
        edge_vectors, radial_emb, senders, receivers, Wr1, Wr2, s_u, v_u, a_s, a_v, E);

    // linear down
    gemm_wmma_f32<<<ggrid, gblk, 0, stream>>>(a_s, C_CH, 1, 0,
                                              W_down_s, C_CH, dn_s, C_CH, 1, 0, N, C_CH);
    gemm_wmma_f32<<<ggridv, gblk, 0, stream>>>(a_v, 3 * C_CH, 3, 1,
                                               W_down_v, C_CH, dn_v, 3 * C_CH, 3, 1, N, C_CH);

    // product basis (elementwise, species-indexed)
    prod_elem<<<(int)((NC + 255) / 256), 256, 0, stream>>>(
        dn_s, dn_v, node_species, w1s, w1v, w2ss, w2vv, w2sv, p_s, p_v, N);

    // product linear
    gemm_wmma_f32<<<ggrid, gblk, 0, stream>>>(p_s, C_CH, 1, 0,
                                              W_prod_s, C_CH, q_s, C_CH, 1, 0, N, C_CH);
    gemm_wmma_f32<<<ggridv, gblk, 0, stream>>>(p_v, 3 * C_CH, 3, 1,
                                               W_prod_v, C_CH, q_v, 3 * C_CH, 3, 1, N, C_CH);

    // soft-norm + residual + readout
    final_kernel<<<N / 4, dim3(C_CH, 4), 0, stream>>>(
        q_s, q_v, res_s, res_v, w_read, out_node, out_fs, out_fv, N);
}